// BinaryContextLayer_59279138619704
// MI455X (gfx1250) — compile-verified
//
#include <hip/hip_runtime.h>
#include <hip/hip_bf16.h>
#include <math.h>

#define FT 8     // feature_type_count (structural constant of the reference)
#define DD 64    // feature dim per edge

// Scalar (SGPR) 8-way update: packed entry is wave-uniform, so this lowers to
// an s_cmp/s_cbranch chain with one v_max_f32 on the taken path.
#define UPDATE(pk, vv)                                   \
  switch ((pk) & 7) {                                    \
    case 0: a0 = fmaxf(a0, (vv)); ++m0; break;           \
    case 1: a1 = fmaxf(a1, (vv)); ++m1; break;           \
    case 2: a2 = fmaxf(a2, (vv)); ++m2; break;           \
    case 3: a3 = fmaxf(a3, (vv)); ++m3; break;           \
    case 4: a4 = fmaxf(a4, (vv)); ++m4; break;           \
    case 5: a5 = fmaxf(a5, (vv)); ++m5; break;           \
    case 6: a6 = fmaxf(a6, (vv)); ++m6; break;           \
    default: a7 = fmaxf(a7, (vv)); ++m7; break;          \
  }

// ---------------------------------------------------------------------------
// Hot kernel: one 64-thread block (2 wave32) per output slot k; lane d owns
// column d. Unroll-4 over the node's CSR segment: 4 uniform list entries merge
// into one s_load_b128, then 4 independent global_load_b32 row fetches are in
// flight before the first v_max consumes them (4x the memory-level parallelism
// of the naive loop).
// ---------------------------------------------------------------------------
__global__ __launch_bounds__(64)
void gather_max_kernel(const float* __restrict__ evals,
                       const int* __restrict__ list,
                       const int* __restrict__ deg,
                       const int* __restrict__ off,
                       const int* __restrict__ core,
                       float* __restrict__ out) {
  const int k    = blockIdx.x;
  const int d    = threadIdx.x;        // 0..63
  const int node = core[k];
  const int tot  = deg[node];

  float a0=-INFINITY,a1=-INFINITY,a2=-INFINITY,a3=-INFINITY,
        a4=-INFINITY,a5=-INFINITY,a6=-INFINITY,a7=-INFINITY;
  int m0=0,m1=0,m2=0,m3=0,m4=0,m5=0,m6=0,m7=0;

  if (tot > 0) {
    const int base = off[node];
    int i = 0;
    for (; i + 4 <= tot; i += 4) {
      const int p0 = __builtin_amdgcn_readfirstlane(list[base + i + 0]);
      const int p1 = __builtin_amdgcn_readfirstlane(list[base + i + 1]);
      const int p2 = __builtin_amdgcn_readfirstlane(list[base + i + 2]);
      const int p3 = __builtin_amdgcn_readfirstlane(list[base + i + 3]);
      const float v0 = evals[(size_t)(p0 >> 3) * DD + d];
      const float v1 = evals[(size_t)(p1 >> 3) * DD + d];
      const float v2 = evals[(size_t)(p2 >> 3) * DD + d];
      const float v3 = evals[(size_t)(p3 >> 3) * DD + d];
      UPDATE(p0, v0);
      UPDATE(p1, v1);
      UPDATE(p2, v2);
      UPDATE(p3, v3);
    }
    for (; i < tot; ++i) {
      const int p = __builtin_amdgcn_readfirstlane(list[base + i]);
      const float v = evals[(size_t)(p >> 3) * DD + d];
      UPDATE(p, v);
    }
  }

  // Reference semantics:
  //   no edges at all             -> 0
  //   all edges match f (mat==tot)-> raw max (finite, may be negative)
  //   otherwise                   -> max(best_matching, 0)  (covers mat==0)
  float* o = out + ((size_t)k * (FT * DD)) + d;
  if (tot == 0) {
    #pragma unroll
    for (int f = 0; f < FT; ++f) o[(size_t)f * DD] = 0.0f;
  } else {
    o[0*DD] = (m0==tot) ? a0 : fmaxf(a0, 0.0f);
    o[1*DD] = (m1==tot) ? a1 : fmaxf(a1, 0.0f);
    o[2*DD] = (m2==tot) ? a2 : fmaxf(a2, 0.0f);
    o[3*DD] = (m3==tot) ? a3 : fmaxf(a3, 0.0f);
    o[4*DD] = (m4==tot) ? a4 : fmaxf(a4, 0.0f);
    o[5*DD] = (m5==tot) ? a5 : fmaxf(a5, 0.0f);
    o[6*DD] = (m6==tot) ? a6 : fmaxf(a6, 0.0f);
    o[7*DD] = (m7==tot) ? a7 : fmaxf(a7, 0.0f);
  }
}

// ---- CSR build passes -----------------------------------------------------

__global__ void zero_i32_kernel(int* __restrict__ p, long long n) {
  long long i = (long long)blockIdx.x * blockDim.x + threadIdx.x;
  if (i < n) p[i] = 0;
}

__global__ void mark_live_kernel(const int* __restrict__ core,
                                 int* __restrict__ live, int K) {
  int k = blockIdx.x * blockDim.x + threadIdx.x;
  if (k < K) live[core[k]] = 1;          // idempotent plain store
}

__global__ void count_deg_kernel(const int* __restrict__ esrc,
                                 const int* __restrict__ live,
                                 int* __restrict__ deg, int n_edges) {
  int e = blockIdx.x * blockDim.x + threadIdx.x;
  if (e >= n_edges) return;
  int s = esrc[e];
  if (live[s]) atomicAdd(&deg[s], 1);    // global_atomic_add_u32, no return
}

__global__ void reserve_kernel(const int* __restrict__ deg,
                               int* __restrict__ off,
                               int* __restrict__ cursor, int n_nodes) {
  int n = blockIdx.x * blockDim.x + threadIdx.x;
  if (n >= n_nodes) return;
  int dg = deg[n];
  if (dg > 0) off[n] = atomicAdd(cursor, dg);
}

__global__ void fill_kernel(const int* __restrict__ esrc,
                            const int* __restrict__ edst,
                            const int* __restrict__ raw,
                            const int* __restrict__ live,
                            const int* __restrict__ off,
                            int* __restrict__ fill,
                            int* __restrict__ list, int n_edges) {
  int e = blockIdx.x * blockDim.x + threadIdx.x;
  if (e >= n_edges) return;
  int s = esrc[e];
  if (!live[s]) return;
  int f = raw[edst[e]] & 7;              // feature type resolved once, here
  int p = off[s] + atomicAdd(&fill[s], 1);
  list[p] = (e << 3) | f;                // e < 2^21, fits with 3 type bits
}

// ---------------------------------------------------------------------------

extern "C" void kernel_launch(void* const* d_in, const int* in_sizes, int n_in,
                              void* d_out, int out_size, void* d_ws, size_t ws_size,
                              hipStream_t stream) {
  const int*   raw   = (const int*)d_in[0];     // raw_data   [N]
  const int*   esrc  = (const int*)d_in[1];     // edge_src   [E]
  const int*   edst  = (const int*)d_in[2];     // edge_dst   [E]
  const float* evals = (const float*)d_in[3];   // edge_vals  [E, 64]
  const int*   core  = (const int*)d_in[4];     // core_idxs  [K]
  // d_in[5] = feature_type_count (device scalar) -- structural constant FT=8.

  const int n_nodes = in_sizes[0];
  const int n_edges = in_sizes[1];
  const int K       = in_sizes[4];

  float* out = (float*)d_out;                   // [K, FT, DD]

  // Workspace layout (ints): [deg N][fill N][live N][cursor 1][off N][list E]
  int* ws    = (int*)d_ws;
  int* deg   = ws;
  int* fillc = ws + n_nodes;
  int* live  = ws + 2 * (size_t)n_nodes;
  int* curs  = ws + 3 * (size_t)n_nodes;
  int* off   = ws + 3 * (size_t)n_nodes + 1;
  int* list  = ws + 4 * (size_t)n_nodes + 1;

  const int T = 256;
  const long long zn = 3LL * n_nodes + 1;       // deg+fill+live+cursor contiguous

  zero_i32_kernel<<<(unsigned)((zn + T - 1) / T), T, 0, stream>>>(deg, zn);
  mark_live_kernel<<<(unsigned)((K + T - 1) / T), T, 0, stream>>>(core, live, K);
  count_deg_kernel<<<(unsigned)((n_edges + T - 1) / T), T, 0, stream>>>(esrc, live, deg, n_edges);
  reserve_kernel<<<(unsigned)((n_nodes + T - 1) / T), T, 0, stream>>>(deg, off, curs, n_nodes);
  fill_kernel<<<(unsigned)((n_edges + T - 1) / T), T, 0, stream>>>(
      esrc, edst, raw, live, off, fillc, list, n_edges);
  gather_max_kernel<<<K, 64, 0, stream>>>(evals, list, deg, off, core, out);
}